// EMContextualCapsuleLayer_10161892623156
// MI455X (gfx1250) — compile-verified
//
#include <hip/hip_runtime.h>
#include <math.h>
#include <stdint.h>

#define EPSF     1e-8f
#define LBDF     0.001f
#define LN2PIF   1.8378770664093453f   // log(2*pi)

typedef float v2f __attribute__((ext_vector_type(2)));
typedef float v8f __attribute__((ext_vector_type(8)));

__device__ __forceinline__ v8f wmma4(v2f a, v2f b, v8f c) {
    // V_WMMA_F32_16X16X4_F32 : D = A(16x4) x B(4x16) + C(16x16), fp32
    return __builtin_amdgcn_wmma_f32_16x16x4_f32(
        false, a, false, b, (short)0, c, false, false);
}

// ---------------------------------------------------------------------------
// actn_in[b,n] = sigmoid(inputs_u[b,n,:] . Wa + ba)
// ---------------------------------------------------------------------------
__global__ void k_actn(const float* __restrict__ xu, const float* __restrict__ Wa,
                       const float* __restrict__ ba, float* __restrict__ actn) {
    int tid = blockIdx.x * blockDim.x + threadIdx.x;
    if (tid < 512) {
        const float* row = xu + (size_t)tid * 128;
        float s = 0.f;
        #pragma unroll 4
        for (int d = 0; d < 128; ++d) s += row[d] * Wa[d];
        s += ba[0];
        actn[tid] = 1.f / (1.f + expf(-s));
    }
}

// ---------------------------------------------------------------------------
// c = context(256x512) @ Wc(512x64) + bc   (one wave per 16x16 tile)
// ---------------------------------------------------------------------------
__global__ void k_c(const float* __restrict__ ctx, const float* __restrict__ Wc,
                    const float* __restrict__ bc, float* __restrict__ cout) {
    int lane = threadIdx.x;
    int mt = blockIdx.x;            // 16 row tiles
    int nt = blockIdx.y;            // 4 col tiles
    int m  = lane & 15;
    int ko = (lane >> 4) << 1;
    v8f acc = {};
    const float* Arow = ctx + (size_t)(mt * 16 + m) * 512;
    const float* Bp   = Wc + nt * 16 + m;
    for (int k = 0; k < 512; k += 4) {
        v2f a = { Arow[k + ko], Arow[k + ko + 1] };
        v2f b = { Bp[(size_t)(k + ko) * 64], Bp[(size_t)(k + ko + 1) * 64] };
        acc = wmma4(a, b, acc);
    }
    int rsh = (lane >> 4) * 8;
    int col = nt * 16 + m;
    for (int v = 0; v < 8; ++v)
        cout[(size_t)(mt * 16 + v + rsh) * 64 + col] = acc[v] + bc[col];
}

// ---------------------------------------------------------------------------
// priors[b,n,o,e] = sum_d inputs_u[b,n,d] * route_weights[n,o,d,e]
// per (n,o,ntile): (8x128)@(128x16) padded to M=16
// ---------------------------------------------------------------------------
__global__ void k_priors(const float* __restrict__ xu, const float* __restrict__ rw,
                         float* __restrict__ priors) {
    int lane = threadIdx.x;
    int n  = blockIdx.x;            // 64
    int o  = blockIdx.y;            // 16
    int nt = blockIdx.z;            // 4
    int m  = lane & 15;
    int ko = (lane >> 4) << 1;
    v8f acc = {};
    const float* Arow = xu + ((size_t)(m < 8 ? m : 0) * 64 + n) * 128;
    const float  msk  = (m < 8) ? 1.f : 0.f;
    const float* Bp   = rw + ((size_t)(n * 16 + o) * 128) * 64 + nt * 16 + m;
    for (int k = 0; k < 128; k += 4) {
        v2f a = { msk * Arow[k + ko], msk * Arow[k + ko + 1] };
        v2f b = { Bp[(size_t)(k + ko) * 64], Bp[(size_t)(k + ko + 1) * 64] };
        acc = wmma4(a, b, acc);
    }
    int rsh = (lane >> 4) * 8;
    for (int v = 0; v < 8; ++v) {
        int bb = v + rsh;           // batch row
        if (bb < 8)
            priors[(((size_t)bb * 64 + n) * 16 + o) * 64 + nt * 16 + m] = acc[v];
    }
}

// ---------------------------------------------------------------------------
// u = priors(8192x64) @ Wu(64x64) + bu
// ---------------------------------------------------------------------------
__global__ void k_u(const float* __restrict__ priors, const float* __restrict__ Wu,
                    const float* __restrict__ bu, float* __restrict__ u) {
    int lane = threadIdx.x;
    int mt = blockIdx.x;            // 512
    int nt = blockIdx.y;            // 4
    int m  = lane & 15;
    int ko = (lane >> 4) << 1;
    v8f acc = {};
    const float* Arow = priors + (size_t)(mt * 16 + m) * 64;
    const float* Bp   = Wu + nt * 16 + m;
    for (int k = 0; k < 64; k += 4) {
        v2f a = { Arow[k + ko], Arow[k + ko + 1] };
        v2f b = { Bp[(size_t)(k + ko) * 64], Bp[(size_t)(k + ko + 1) * 64] };
        acc = wmma4(a, b, acc);
    }
    int rsh = (lane >> 4) * 8;
    int col = nt * 16 + m;
    for (int v = 0; v < 8; ++v)
        u[(size_t)(mt * 16 + v + rsh) * 64 + col] = acc[v] + bu[col];
}

// ---------------------------------------------------------------------------
// Fused: votes = tanh(u[b] + c[b,l]) @ Wv + bv  (kept entirely in LDS, 256KB)
// then 3-iteration EM routing. One workgroup (8 waves) per (b,l).
// Wv (16KB) is staged into LDS via GLOBAL_LOAD_ASYNC_TO_LDS_B128 (ASYNCcnt).
// Dynamic LDS: 74032 floats = 296128 bytes (< 320KB WGP LDS).
// ---------------------------------------------------------------------------
__global__ void __launch_bounds__(256) k_votes_em(
    const float* __restrict__ u, const float* __restrict__ cmat,
    const float* __restrict__ Wv, const float* __restrict__ bv,
    const float* __restrict__ actn, const unsigned char* __restrict__ maskb,
    const float* __restrict__ beta_u, const float* __restrict__ beta_a,
    float* __restrict__ out_mu, float* __restrict__ out_r)
{
    extern __shared__ float smem[];
    float* votes  = smem;                 // 65536 : [1024 rows (n,o)][64 e]
    float* rbuf   = smem + 65536;         // 1024  : r[n][o]
    float* r1     = rbuf + 1024;          // 1024
    float* mu     = r1 + 1024;            // 1024  : [o][e]
    float* sig    = mu + 1024;            // 1024
    float* rsum   = sig + 1024;           // 64
    float* arsum  = rsum + 64;            // 16
    float* csum   = arsum + 16;           // 16 : sum_e log(sigma+eps)
    float* lact   = csum + 16;            // 16 : log(actn_out)
    float* actn_s = lact + 16;            // 64
    float* maskv  = actn_s + 64;          // 64
    float* crow   = maskv + 64;           // 64
    float* wv_s   = crow + 64;            // 4096 : Wv staged in LDS

    int tid  = threadIdx.x;
    int lane = tid & 31;
    int wave = tid >> 5;
    int bl = blockIdx.x;                  // 0..255  (b*32 + l)
    int b  = bl >> 5;

    // ---- async DMA of Wv (64x64 f32 = 16KB) global -> LDS, 16B per copy ----
    {
        unsigned ldsbase = (unsigned)(unsigned long long)(uintptr_t)(void*)wv_s;
        unsigned long long gbase = (unsigned long long)(uintptr_t)Wv;
        #pragma unroll
        for (int i = 0; i < 4; ++i) {
            unsigned           laddr = ldsbase + (unsigned)(tid + i * 256) * 16u;
            unsigned long long gaddr = gbase + (unsigned long long)(tid + i * 256) * 16ull;
            asm volatile("global_load_async_to_lds_b128 %0, %1, off"
                         :: "v"(laddr), "v"(gaddr) : "memory");
        }
        asm volatile("s_wait_asynccnt 0x0" ::: "memory");
    }

    if (tid < 64) {
        crow[tid]   = cmat[(size_t)bl * 64 + tid];
        actn_s[tid] = actn[b * 64 + tid];
        maskv[tid]  = maskb[b * 64 + tid] ? 1.f : 0.f;
    }
    for (int i = tid; i < 1024; i += 256) rbuf[i] = 1.f / 16.f;
    __syncthreads();

    // ---- votes via WMMA (computed transposed: D = Wv^T x T^T) ----
    const float* ub = u + (size_t)b * 1024 * 64;
    int m16 = lane & 15;
    int ko  = (lane >> 4) << 1;
    int rsh = (lane >> 4) * 8;
    for (int rt = wave; rt < 64; rt += 8) {
        int R = rt * 16 + m16;            // flattened (n,o) row
        const float* urow = ub + (size_t)R * 64;
        v2f bf[16];                       // tanh(u+c): each element computed once
        #pragma unroll
        for (int ks = 0; ks < 16; ++ks) {
            int k = ks * 4 + ko;
            bf[ks].x = tanhf(urow[k]     + crow[k]);
            bf[ks].y = tanhf(urow[k + 1] + crow[k + 1]);
        }
        for (int et = 0; et < 4; ++et) {
            v8f acc = {};
            #pragma unroll
            for (int ks = 0; ks < 16; ++ks) {
                int k = ks * 4 + ko;
                v2f a = { wv_s[k * 64 + et * 16 + m16],
                          wv_s[(k + 1) * 64 + et * 16 + m16] };
                acc = wmma4(a, bf[ks], acc);
            }
            for (int v = 0; v < 8; ++v) {
                int e   = et * 16 + v + rsh;     // D row = e index
                int row = rt * 16 + m16;         // D col = (n,o) row
                votes[(size_t)row * 64 + e] = acc[v] + bv[e];
            }
        }
    }
    __syncthreads();

    // ---- EM routing (3 iters) ----
    for (int it = 0; it < 3; ++it) {
        if (tid < 64) {                   // sum_o r[n][o]
            float s = 0.f;
            for (int o = 0; o < 16; ++o) s += rbuf[tid * 16 + o];
            rsum[tid] = s;
        }
        __syncthreads();
        for (int i = tid; i < 1024; i += 256) {   // ar normalized over o
            int n = i >> 4;
            float a = actn_s[n];
            r1[i] = a * rbuf[i] / (a * rsum[n] + EPSF);
        }
        __syncthreads();
        if (tid < 16) {                   // ar_sum[o] = sum_n ar
            float s = 0.f;
            for (int n = 0; n < 64; ++n) s += r1[n * 16 + tid];
            arsum[tid] = s;
        }
        __syncthreads();
        for (int i = tid; i < 1024; i += 256)     // r1 = ar/(ar_sum+eps)
            r1[i] = r1[i] / (arsum[i & 15] + EPSF);
        __syncthreads();
        for (int i = tid; i < 1024; i += 256) {   // mu[o][e]
            int o = i >> 6, e = i & 63;
            float s = 0.f;
            for (int n = 0; n < 64; ++n)
                s += r1[n * 16 + o] * votes[(size_t)((n * 16 + o) << 6) + e];
            mu[i] = s;
        }
        __syncthreads();
        for (int i = tid; i < 1024; i += 256) {   // sigma[o][e]
            int o = i >> 6, e = i & 63;
            float m = mu[i], s = 0.f;
            for (int n = 0; n < 64; ++n) {
                float d = votes[(size_t)((n * 16 + o) << 6) + e] - m;
                s += r1[n * 16 + o] * d * d;
            }
            sig[i] = s;
        }
        __syncthreads();
        if (tid < 16) {                   // cost -> actn_out
            float cl = 0.f;
            for (int e = 0; e < 64; ++e) cl += logf(sig[tid * 64 + e] + EPSF);
            csum[tid] = cl;
            float cost = 64.f * beta_u[tid] + 0.5f * cl * arsum[tid];
            float ao = 1.f / (1.f + expf(-LBDF * (beta_a[tid] - cost)));
            lact[tid] = logf(ao);
        }
        __syncthreads();
        if (it < 2) {                     // e-step
            for (int i = tid; i < 1024; i += 256) {
                int n = i >> 4, o = i & 15;
                const float* vp = votes + (size_t)((n * 16 + o) << 6);
                const float* mp = mu + (o << 6);
                const float* sp = sig + (o << 6);
                float s = 0.f;
                for (int e = 0; e < 64; ++e) {
                    float d = vp[e] - mp[e];
                    s += d * d / (sp[e] + EPSF);
                }
                float lp = -0.5f * s - 0.5f * csum[o] - 32.f * LN2PIF + lact[o];
                rbuf[i] = (maskv[n] != 0.f) ? -10000.f : lp;
            }
            __syncthreads();
            if (tid < 64) {               // softmax over o
                float mx = -1e30f;
                for (int o = 0; o < 16; ++o) mx = fmaxf(mx, rbuf[tid * 16 + o]);
                float ex[16], ss = 0.f;
                for (int o = 0; o < 16; ++o) {
                    ex[o] = expf(rbuf[tid * 16 + o] - mx);
                    ss += ex[o];
                }
                for (int o = 0; o < 16; ++o) rbuf[tid * 16 + o] = ex[o] / ss;
            }
            __syncthreads();
        }
    }

    for (int i = tid; i < 1024; i += 256) {
        out_mu[(size_t)bl * 1024 + i] = mu[i];
        out_r [(size_t)bl * 1024 + i] = rbuf[i];
    }
}

// ---------------------------------------------------------------------------
extern "C" void kernel_launch(void* const* d_in, const int* in_sizes, int n_in,
                              void* d_out, int out_size, void* d_ws, size_t ws_size,
                              hipStream_t stream) {
    const float* xu   = (const float*)d_in[0];
    const unsigned char* mask = (const unsigned char*)d_in[1];
    const float* ctx  = (const float*)d_in[2];
    const float* rw   = (const float*)d_in[3];
    const float* Wu   = (const float*)d_in[4];
    const float* bu   = (const float*)d_in[5];
    const float* Wc   = (const float*)d_in[6];
    const float* bc   = (const float*)d_in[7];
    const float* Wv   = (const float*)d_in[8];
    const float* bv   = (const float*)d_in[9];
    const float* Wa   = (const float*)d_in[10];
    const float* ba   = (const float*)d_in[11];
    const float* beta_u = (const float*)d_in[12];
    const float* beta_a = (const float*)d_in[13];
    float* out = (float*)d_out;

    float* ws      = (float*)d_ws;
    float* ws_actn = ws;                       // 512
    float* ws_c    = ws + 512;                 // 16384
    float* ws_pri  = ws + 512 + 16384;         // 524288
    float* ws_u    = ws_pri + 524288;          // 524288

    k_actn  <<<2, 256, 0, stream>>>(xu, Wa, ba, ws_actn);
    k_c     <<<dim3(16, 4), 32, 0, stream>>>(ctx, Wc, bc, ws_c);
    k_priors<<<dim3(64, 16, 4), 32, 0, stream>>>(xu, rw, ws_pri);
    k_u     <<<dim3(512, 4), 32, 0, stream>>>(ws_pri, Wu, bu, ws_u);

    size_t shbytes = (size_t)(65536 + 4 * 1024 + 64 + 16 + 16 + 16 + 64 + 64 + 64
                              + 4096) * sizeof(float);   // 296128 B
    k_votes_em<<<256, 256, shbytes, stream>>>(ws_u, ws_c, Wv, bv, ws_actn, mask,
                                              beta_u, beta_a,
                                              out, out + 262144);
}